// Decoder_90615220011707
// MI455X (gfx1250) — compile-verified
//
#include <hip/hip_runtime.h>
#include <hip/hip_bf16.h>
#include <math.h>
#include <stdint.h>

// Problem constants (from reference)
#define HIDDEN 1024
#define VOCAB  32000
#define BATCH  16
#define TSTEPS 11
#define SOS    1
#define GATES  (4 * HIDDEN)

// TDM tiling for the logits GEMM (2D tiles)
#define KBLK   128                 // K-chunk staged per TDM transfer
#define KPAD   4                   // dwords of pad per row (bank-conflict-free)
#define ROWF   (KBLK + KPAD)       // LDS row pitch in floats (132)
#define NCHUNK (HIDDEN / KBLK)     // 8

// TDM tiling for the gates GEMM (3D tiles: K x 16 rows x 4 gate blocks)
#define GK     64                  // K-chunk per TDM transfer
#define GROWF  (GK + KPAD)         // LDS row pitch (68)
#define GNCH   (HIDDEN / GK)       // 16

typedef __attribute__((ext_vector_type(2))) float v2f;
typedef __attribute__((ext_vector_type(8))) float v8f;
typedef __attribute__((ext_vector_type(4))) unsigned int u32x4;
typedef __attribute__((ext_vector_type(4))) int i32x4;
typedef __attribute__((ext_vector_type(8))) int i32x8;

// D(16x16,f32) += A(16x4,f32) * B(4x16,f32) ; full-precision CDNA5 WMMA.
__device__ __forceinline__ v8f wmma4(v2f a, v2f b, v8f c) {
  return __builtin_amdgcn_wmma_f32_16x16x4_f32(
      /*neg_a=*/false, a, /*neg_b=*/false, b,
      /*c_mod=*/(short)0, c, /*reuse_a=*/false, /*reuse_b=*/false);
}

__device__ __forceinline__ float sigmoid_f(float x) {
  return 1.0f / (1.0f + __expf(-x));
}

// ---------------------------------------------------------------------------
// Generalized TDM descriptor: stage a [tileZ x tileRows x tileK] f32 tile of a
// row-major matrix (row stride rowStride elems, z stride zStride elems) into
// LDS at ldsaddr, padding 4 dwords after each tileK-dword row.
// D# layout per CDNA5 ISA.  tileZ==0 -> plain 2D tile.
// ---------------------------------------------------------------------------
__device__ __forceinline__ void tdm_load(uint64_t gaddr, uint32_t ldsaddr,
                                         uint32_t tileK, uint32_t padCode,
                                         uint32_t tileRows, uint32_t tileZ,
                                         uint32_t nRowsTotal,
                                         uint64_t rowStride, uint64_t zStride,
                                         uint32_t tensorDim2) {
  u32x4 g0;
  g0[0] = 1u;                                       // count=1, user descriptor
  g0[1] = ldsaddr;                                  // lds_addr [63:32]
  g0[2] = (uint32_t)gaddr;                          // global_addr [95:64]
  g0[3] = (uint32_t)((gaddr >> 32) & 0x01FFFFFFu)   // global_addr [120:96]
          | (2u << 30);                             // type=2 ("image")

  i32x8 g1;
  g1[0] = (int)((2u << 16)                // data_size = 4B
              | (1u << 20)                // pad_enable
              | (padCode << 22)           // pad_interval (dword count code)
              | (3u << 25));              // pad_amount: 4 dwords
  g1[1] = (int)(((uint32_t)HIDDEN & 0xFFFFu) << 16);            // tensor_dim0 lo
  g1[2] = (int)((((uint32_t)HIDDEN >> 16) & 0xFFFFu)
              | ((nRowsTotal & 0xFFFFu) << 16));                // dim0 hi/dim1 lo
  g1[3] = (int)(((nRowsTotal >> 16) & 0xFFFFu) | (tileK << 16));// dim1 hi/tile_dim0
  g1[4] = (int)(tileRows | (tileZ << 16));          // tile_dim1 | tile_dim2
  g1[5] = (int)(uint32_t)rowStride;                 // tensor_dim0_stride lo32
  g1[6] = (int)((uint32_t)((rowStride >> 32) & 0xFFFFu)
              | ((uint32_t)(zStride & 0xFFFFu) << 16));         // d0s hi|d1s lo
  g1[7] = (int)(uint32_t)(zStride >> 16);           // tensor_dim1_stride [47:16]

  i32x4 g2 = {(int)tensorDim2, 0, 0, 0};            // tensor_dim2 (3D OOB bound)
  i32x4 g3 = {0, 0, 0, 0};
#if defined(__clang_major__) && __clang_major__ >= 23
  i32x8 z8 = {0, 0, 0, 0, 0, 0, 0, 0};
  __builtin_amdgcn_tensor_load_to_lds(g0, g1, g2, g3, z8, 0);
#else
  __builtin_amdgcn_tensor_load_to_lds(g0, g1, g2, g3, 0);
#endif
}

// ---------------------------------------------------------------------------
// hh[16,4096] = h0[16,1024] @ W_hh^T + (b_ih + b_hh)      (run ONCE)
// One wave per 16-column tile; 256 waves -> 32 blocks x 256 threads.
// ---------------------------------------------------------------------------
__global__ void hh_precompute(const float* __restrict__ h0,
                              const float* __restrict__ W_hh,
                              const float* __restrict__ b_ih,
                              const float* __restrict__ b_hh,
                              float* __restrict__ hh) {
  const int wave = (blockIdx.x * blockDim.x + threadIdx.x) >> 5;
  const int n0   = wave * 16;
  const int lane = threadIdx.x & 31;
  const int n    = lane & 15;
  const int half = lane >> 4;
  const int kk   = half << 1;

  const float bias = b_ih[n0 + n] + b_hh[n0 + n];
  v8f acc;
#pragma unroll
  for (int r = 0; r < 8; ++r) acc[r] = bias;

  const float* aRow = h0 + (size_t)n * HIDDEN + kk;
  const float* wRow = W_hh + (size_t)(n0 + n) * HIDDEN + kk;
  for (int k = 0; k < HIDDEN; k += 4)
    acc = wmma4(*(const v2f*)(aRow + k), *(const v2f*)(wRow + k), acc);

#pragma unroll
  for (int r = 0; r < 8; ++r) {
    const int m = r + 8 * half;
    hh[(size_t)m * GATES + n0 + n] = acc[r];
  }
}

// ---------------------------------------------------------------------------
// x[16,1024] = relu(emb[tok])        (per step; 64 blocks x 256)
// ---------------------------------------------------------------------------
__global__ void embed_relu(const float* __restrict__ emb,
                           const int* __restrict__ tok, int use_sos,
                           float* __restrict__ x) {
  const int i = blockIdx.x * blockDim.x + threadIdx.x;
  const int b = i >> 10;
  const int k = i & (HIDDEN - 1);
  const int tkn = use_sos ? SOS : tok[b];
  const float v = emb[(size_t)tkn * HIDDEN + k];
  x[i] = fmaxf(v, 0.0f);
}

// ---------------------------------------------------------------------------
// Fused gates GEMM + LSTM cell, TDM-staged.
// One wave per 16-column tile (64 blocks x 32 threads). A single 3D TDM
// descriptor per K-chunk fetches the i/f/g/o row blocks (tile_dim2=4, z-stride
// HIDDEN*HIDDEN) into a 68-float-pitch LDS image; double-buffered.
// ---------------------------------------------------------------------------
__global__ void __launch_bounds__(32)
gates_cell_tdm(const float* __restrict__ x,
               const float* __restrict__ W_ih,
               const float* __restrict__ hh,
               const float* __restrict__ c0,
               float* __restrict__ h) {
  __shared__ __align__(16) float smem[2 /*bufs*/ * 4 /*gates*/ * 16 * GROWF];

  const int j0   = blockIdx.x * 16;
  const int lane = threadIdx.x & 31;
  const int n    = lane & 15;
  const int half = lane >> 4;
  const int kk   = half << 1;

  float* buf0 = smem;
  float* buf1 = smem + 4 * 16 * GROWF;
  const uint32_t lds0 = (uint32_t)(uintptr_t)(void*)buf0;
  const uint32_t lds1 = (uint32_t)(uintptr_t)(void*)buf1;
  const uint64_t wbase =
      (uint64_t)(uintptr_t)W_ih + (uint64_t)j0 * HIDDEN * sizeof(float);

  // Prime double buffer: chunks 0 and 1 in flight.
  tdm_load(wbase + (uint64_t)0 * GK * sizeof(float), lds0, GK, /*pad=64dw*/5,
           16, 4, GATES, HIDDEN, (uint64_t)HIDDEN * HIDDEN, 4);
  tdm_load(wbase + (uint64_t)1 * GK * sizeof(float), lds1, GK, 5,
           16, 4, GATES, HIDDEN, (uint64_t)HIDDEN * HIDDEN, 4);

  v8f ai, af, ag, ao;
#pragma unroll
  for (int r = 0; r < 8; ++r) {
    const int m = r + 8 * half;
    const float* hr = hh + (size_t)m * GATES + j0 + n;
    ai[r] = hr[0 * HIDDEN];
    af[r] = hr[1 * HIDDEN];
    ag[r] = hr[2 * HIDDEN];
    ao[r] = hr[3 * HIDDEN];
  }

  const float* aRow = x + (size_t)n * HIDDEN + kk;

  for (int c = 0; c < GNCH; ++c) {
    if (c == GNCH - 1) __builtin_amdgcn_s_wait_tensorcnt(0);
    else               __builtin_amdgcn_s_wait_tensorcnt(1);
    asm volatile("" ::: "memory");  // DMA data visible; no ds hoisting

    const float* base = (c & 1) ? buf1 : buf0;
    const float* bI = base + ((size_t)0 * 16 + n) * GROWF + kk;
    const float* bF = base + ((size_t)1 * 16 + n) * GROWF + kk;
    const float* bG = base + ((size_t)2 * 16 + n) * GROWF + kk;
    const float* bO = base + ((size_t)3 * 16 + n) * GROWF + kk;
    const float* aC = aRow + c * GK;
#pragma unroll
    for (int k4 = 0; k4 < GK; k4 += 4) {
      const v2f a = *(const v2f*)(aC + k4);
      ai = wmma4(a, *(const v2f*)(bI + k4), ai);
      af = wmma4(a, *(const v2f*)(bF + k4), af);
      ag = wmma4(a, *(const v2f*)(bG + k4), ag);
      ao = wmma4(a, *(const v2f*)(bO + k4), ao);
    }

    if (c + 2 < GNCH) {
      asm volatile("s_wait_dscnt 0" ::: "memory");  // buffer fully consumed
      tdm_load(wbase + (uint64_t)(c + 2) * GK * sizeof(float),
               (c & 1) ? lds1 : lds0, GK, 5,
               16, 4, GATES, HIDDEN, (uint64_t)HIDDEN * HIDDEN, 4);
    }
  }

#pragma unroll
  for (int r = 0; r < 8; ++r) {
    const int m = r + 8 * half;
    const float iv = sigmoid_f(ai[r]);
    const float fv = sigmoid_f(af[r]);
    const float gv = tanhf(ag[r]);
    const float ov = sigmoid_f(ao[r]);
    const float cv = fv * c0[(size_t)m * HIDDEN + j0 + n] + iv * gv;
    h[(size_t)m * HIDDEN + j0 + n] = ov * tanhf(cv);
  }
}

// ---------------------------------------------------------------------------
// logits[16,32000] = h @ fc_W^T + fc_b  ->  d_out[b][t][v]
// TDM-staged, double-buffered. 2 waves/block (64 thr), 1000 blocks.
// ---------------------------------------------------------------------------
__global__ void __launch_bounds__(64)
logits_gemm_tdm(const float* __restrict__ hmat,
                const float* __restrict__ fc_W,
                const float* __restrict__ fc_b,
                float* __restrict__ out, int t) {
  __shared__ __align__(16) float smem[2 /*waves*/ * 2 /*bufs*/ * 16 * ROWF];

  const int wib  = __builtin_amdgcn_readfirstlane((int)(threadIdx.x >> 5));
  const int wave = blockIdx.x * 2 + wib;
  const int n0   = wave * 16;
  const int lane = threadIdx.x & 31;
  const int n    = lane & 15;
  const int half = lane >> 4;
  const int kk   = half << 1;

  float* buf0 = smem + (size_t)wib * (2 * 16 * ROWF);
  float* buf1 = buf0 + 16 * ROWF;
  const uint32_t lds0 = (uint32_t)(uintptr_t)(void*)buf0;
  const uint32_t lds1 = (uint32_t)(uintptr_t)(void*)buf1;

  const uint64_t wbase =
      (uint64_t)(uintptr_t)fc_W + (uint64_t)n0 * HIDDEN * sizeof(float);

  tdm_load(wbase + (uint64_t)0 * KBLK * sizeof(float), lds0, KBLK,
           /*pad=128dw*/6, 16, 0, VOCAB, HIDDEN, 0, 0);
  tdm_load(wbase + (uint64_t)1 * KBLK * sizeof(float), lds1, KBLK,
           6, 16, 0, VOCAB, HIDDEN, 0, 0);

  const float bias = fc_b[n0 + n];
  v8f acc;
#pragma unroll
  for (int r = 0; r < 8; ++r) acc[r] = bias;

  const float* aRow = hmat + (size_t)n * HIDDEN + kk;

  for (int c = 0; c < NCHUNK; ++c) {
    if (c == NCHUNK - 1) __builtin_amdgcn_s_wait_tensorcnt(0);
    else                 __builtin_amdgcn_s_wait_tensorcnt(1);
    asm volatile("" ::: "memory");

    const float* bRow = ((c & 1) ? buf1 : buf0) + (size_t)n * ROWF + kk;
    const float* aC   = aRow + c * KBLK;
#pragma unroll
    for (int k4 = 0; k4 < KBLK; k4 += 4)
      acc = wmma4(*(const v2f*)(aC + k4), *(const v2f*)(bRow + k4), acc);

    if (c + 2 < NCHUNK) {
      asm volatile("s_wait_dscnt 0" ::: "memory");
      tdm_load(wbase + (uint64_t)(c + 2) * KBLK * sizeof(float),
               (c & 1) ? lds1 : lds0, KBLK, 6, 16, 0, VOCAB, HIDDEN, 0, 0);
    }
  }

#pragma unroll
  for (int r = 0; r < 8; ++r) {
    const int m = r + 8 * half;
    out[((size_t)m * TSTEPS + t) * VOCAB + n0 + n] = acc[r];
  }
}

// ---------------------------------------------------------------------------
// Greedy argmax per batch row (first-index tie break). One block per row.
// ---------------------------------------------------------------------------
__global__ void argmax_row(const float* __restrict__ out, int t,
                           int* __restrict__ tok) {
  __shared__ float sv[256];
  __shared__ int   si[256];
  const int b = blockIdx.x;
  const float* row = out + ((size_t)b * TSTEPS + t) * VOCAB;

  float best = -INFINITY;
  int   bidx = 0;
  for (int i = threadIdx.x; i < VOCAB; i += 256) {
    const float v = row[i];
    if (v > best) { best = v; bidx = i; }
  }
  sv[threadIdx.x] = best;
  si[threadIdx.x] = bidx;
  __syncthreads();
  for (int s = 128; s > 0; s >>= 1) {
    if (threadIdx.x < s) {
      const float ov = sv[threadIdx.x + s];
      const int   oi = si[threadIdx.x + s];
      if (ov > sv[threadIdx.x] ||
          (ov == sv[threadIdx.x] && oi < si[threadIdx.x])) {
        sv[threadIdx.x] = ov;
        si[threadIdx.x] = oi;
      }
    }
    __syncthreads();
  }
  if (threadIdx.x == 0) tok[b] = si[0];
}

// ---------------------------------------------------------------------------
// In-place log_softmax over each [VOCAB] row of d_out. 176 blocks.
// ---------------------------------------------------------------------------
__global__ void log_softmax_rows(float* __restrict__ out) {
  __shared__ float red[256];
  float* row = out + (size_t)blockIdx.x * VOCAB;

  float mx = -INFINITY;
  for (int i = threadIdx.x; i < VOCAB; i += 256) mx = fmaxf(mx, row[i]);
  red[threadIdx.x] = mx;
  __syncthreads();
  for (int s = 128; s > 0; s >>= 1) {
    if (threadIdx.x < s)
      red[threadIdx.x] = fmaxf(red[threadIdx.x], red[threadIdx.x + s]);
    __syncthreads();
  }
  mx = red[0];
  __syncthreads();

  float sum = 0.0f;
  for (int i = threadIdx.x; i < VOCAB; i += 256) sum += __expf(row[i] - mx);
  red[threadIdx.x] = sum;
  __syncthreads();
  for (int s = 128; s > 0; s >>= 1) {
    if (threadIdx.x < s) red[threadIdx.x] += red[threadIdx.x + s];
    __syncthreads();
  }
  const float lse = mx + __logf(red[0]);

  for (int i = threadIdx.x; i < VOCAB; i += 256) row[i] -= lse;
}

// ---------------------------------------------------------------------------
// Host launch. Inputs (setup_inputs order):
//  0: encoder_outputs (UNUSED by the reference math)
//  1: h0  2: c0  3: emb  4: W_ih  5: W_hh  6: b_ih  7: b_hh  8: fc_W  9: fc_b
// ---------------------------------------------------------------------------
extern "C" void kernel_launch(void* const* d_in, const int* in_sizes, int n_in,
                              void* d_out, int out_size, void* d_ws, size_t ws_size,
                              hipStream_t stream) {
  (void)in_sizes; (void)n_in; (void)out_size; (void)ws_size;

  const float* h0   = (const float*)d_in[1];
  const float* c0   = (const float*)d_in[2];
  const float* emb  = (const float*)d_in[3];
  const float* W_ih = (const float*)d_in[4];
  const float* W_hh = (const float*)d_in[5];
  const float* b_ih = (const float*)d_in[6];
  const float* b_hh = (const float*)d_in[7];
  const float* fc_W = (const float*)d_in[8];
  const float* fc_b = (const float*)d_in[9];
  float* out = (float*)d_out;

  // Workspace layout (floats): hh[16*4096] | x[16*1024] | h[16*1024] | tok[16]
  float* ws  = (float*)d_ws;
  float* hh  = ws;
  float* x   = ws + (size_t)BATCH * GATES;
  float* hbf = x + (size_t)BATCH * HIDDEN;
  int*   tok = (int*)(hbf + (size_t)BATCH * HIDDEN);

  // Loop-invariant recurrent term + both biases, computed once.
  hh_precompute<<<32, 256, 0, stream>>>(h0, W_hh, b_ih, b_hh, hh);

  for (int t = 0; t < TSTEPS; ++t) {
    embed_relu<<<64, 256, 0, stream>>>(emb, tok, t == 0 ? 1 : 0, x);
    gates_cell_tdm<<<HIDDEN / 16, 32, 0, stream>>>(x, W_ih, hh, c0, hbf);
    logits_gemm_tdm<<<1000, 64, 0, stream>>>(hbf, fc_W, fc_b, out, t);
    if (t + 1 < TSTEPS)
      argmax_row<<<BATCH, 256, 0, stream>>>(out, t, tok);
  }
  log_softmax_rows<<<BATCH * TSTEPS, 256, 0, stream>>>(out);
}